// GAT_16243566314139
// MI455X (gfx1250) — compile-verified
//
#include <hip/hip_runtime.h>
#include <hip/hip_bf16.h>
#include <math.h>

typedef float v2f __attribute__((ext_vector_type(2)));
typedef float v8f __attribute__((ext_vector_type(8)));

#define NEG_SLOPE 0.2f

// ---------------------------------------------------------------------------
// 1) h = x @ W  with fp32 WMMA 16x16x4.
//    Block = 256 threads = 8 waves. Block handles rows [m0, m0+16); wave w
//    handles output cols [16w, 16w+16). A-panel (16 x IN) staged in LDS.
// ---------------------------------------------------------------------------
__launch_bounds__(256)
__global__ void k_gemm(const float* __restrict__ x, const float* __restrict__ W,
                       float* __restrict__ h, int N, int IN, int OUT) {
    // IN <= 256 assumed (fixed problem: IN = 256). Pad stride by 4 floats so
    // lanes 0..15 (stride-IN row accesses) land in distinct LDS banks.
    __shared__ float ldsA[16 * 260];
    const int LSTR = IN + 4;
    const int m0 = blockIdx.x * 16;
    const int t  = threadIdx.x;

    // 16 consecutive rows of x are contiguous in memory -> fully coalesced copy
    const float* __restrict__ xblk = x + (size_t)m0 * IN;
    const size_t xmax = (size_t)N * IN;
    for (int idx = t; idx < 16 * IN; idx += 256) {
        int r = idx / IN, c = idx - r * IN;
        float v = 0.0f;
        if ((size_t)m0 * IN + idx < xmax) v = xblk[idx];
        ldsA[r * LSTR + c] = v;
    }
    __syncthreads();

    const int wave  = t >> 5;        // 0..7 -> output col tile
    const int lane  = t & 31;
    const int n0    = wave * 16;
    const int ncol  = n0 + (lane & 15);
    const int khalf = (lane >> 4) * 2;   // lanes 0-15: K={0,1}; lanes 16-31: K={2,3}
    const int mrow  = lane & 15;

    v8f c = {0.f, 0.f, 0.f, 0.f, 0.f, 0.f, 0.f, 0.f};
    for (int k0 = 0; k0 < IN; k0 += 4) {
        v2f a, b;
        // A 16x4 fp32 layout (ISA 7.12.2): lane m holds (K=khalf, K=khalf+1)
        a.x = ldsA[mrow * LSTR + k0 + khalf];
        a.y = ldsA[mrow * LSTR + k0 + khalf + 1];
        // B 4x16 fp32: row K striped across lanes (lanes 0-15: K=khalf rows)
        b.x = W[(size_t)(k0 + khalf)     * OUT + ncol];
        b.y = W[(size_t)(k0 + khalf + 1) * OUT + ncol];
        c = __builtin_amdgcn_wmma_f32_16x16x4_f32(false, a, false, b,
                                                  (short)0, c, false, false);
    }

    // C/D 16x16 fp32 layout: VGPR v holds row v (lanes 0-15) / row v+8 (16-31)
    const int mbase = m0 + (lane >> 4) * 8;
    #pragma unroll
    for (int v = 0; v < 8; ++v) {
        int m = mbase + v;
        if (m < N) h[(size_t)m * OUT + ncol] = c[v];
    }
}

// ---------------------------------------------------------------------------
// 2) Per-node attention logits a_s = h.att_src, a_d = h.att_dst (wave/node)
// ---------------------------------------------------------------------------
__launch_bounds__(256)
__global__ void k_att(const float* __restrict__ h, const float* __restrict__ att_src,
                      const float* __restrict__ att_dst, float* __restrict__ a_s,
                      float* __restrict__ a_d, int N, int OUT) {
    int gw   = (blockIdx.x * blockDim.x + threadIdx.x) >> 5;
    int lane = threadIdx.x & 31;
    if (gw >= N) return;
    const float* __restrict__ hr = h + (size_t)gw * OUT;
    float ss = 0.f, sd = 0.f;
    for (int k = lane; k < OUT; k += 32) {
        float hv = hr[k];
        ss += hv * att_src[k];
        sd += hv * att_dst[k];
    }
    #pragma unroll
    for (int o = 16; o > 0; o >>= 1) {
        ss += __shfl_xor(ss, o, 32);
        sd += __shfl_xor(sd, o, 32);
    }
    if (lane == 0) { a_s[gw] = ss; a_d[gw] = sd; }
}

// ---------------------------------------------------------------------------
// 3) Init scratch for the (filtered) softmax/scatter phase
// ---------------------------------------------------------------------------
__global__ void k_init(unsigned* __restrict__ mb, float* __restrict__ z,
                       float* __restrict__ oacc, int* __restrict__ cnt,
                       int item_len, int OUT) {
    int t = blockIdx.x * blockDim.x + threadIdx.x;
    if (t == 0) *cnt = 0;
    if (t < item_len) { mb[t] = 0u; z[t] = 0.f; }
    if (t < item_len * OUT) oacc[t] = 0.f;
}

// ---------------------------------------------------------------------------
// 4) Compact edges whose destination contributes to the output (dst<item_len)
//    Candidates: E real edges + item_len self-loops.
// ---------------------------------------------------------------------------
__global__ void k_compact(const long long* __restrict__ ei, int E, int item_len,
                          int M, int2* __restrict__ edges, int* __restrict__ cnt) {
    int e = blockIdx.x * blockDim.x + threadIdx.x;
    if (e >= M) return;
    int s, d;
    if (e < E) { s = (int)ei[e]; d = (int)ei[(size_t)E + e]; }
    else       { s = d = e - E; }                 // self loop
    if (d < item_len) {
        int idx = atomicAdd(cnt, 1);
        edges[idx] = make_int2(s, d);
    }
}

__device__ __forceinline__ unsigned f2mono(float f) {
    unsigned b = __float_as_uint(f);
    return (b & 0x80000000u) ? ~b : (b | 0x80000000u);
}
__device__ __forceinline__ float mono2f(unsigned u) {
    return __uint_as_float((u & 0x80000000u) ? (u & 0x7FFFFFFFu) : ~u);
}

// ---------------------------------------------------------------------------
// 5) Segment max of leaky_relu logits (monotone-uint atomicMax)
// ---------------------------------------------------------------------------
__global__ void k_max(const int2* __restrict__ edges, const int* __restrict__ cnt,
                      const float* __restrict__ a_s, const float* __restrict__ a_d,
                      float* __restrict__ ev, unsigned* __restrict__ mb) {
    int t = blockIdx.x * blockDim.x + threadIdx.x;
    if (t >= *cnt) return;
    int2 e = edges[t];
    float v = a_s[e.x] + a_d[e.y];
    v = v > 0.f ? v : NEG_SLOPE * v;
    ev[t] = v;
    atomicMax(&mb[e.y], f2mono(v));
}

// ---------------------------------------------------------------------------
// 6) p = exp(e - m[dst]); z[dst] += p
// ---------------------------------------------------------------------------
__global__ void k_expsum(const int2* __restrict__ edges, const int* __restrict__ cnt,
                         float* __restrict__ ev, const unsigned* __restrict__ mb,
                         float* __restrict__ z) {
    int t = blockIdx.x * blockDim.x + threadIdx.x;
    if (t >= *cnt) return;
    int2 e = edges[t];
    float p = expf(ev[t] - mono2f(mb[e.y]));
    ev[t] = p;
    atomicAdd(&z[e.y], p);
}

// ---------------------------------------------------------------------------
// 7) out[dst] += (p/z[dst]) * h[src]   (wave per edge, lane-strided features)
// ---------------------------------------------------------------------------
__launch_bounds__(256)
__global__ void k_scatter(const int2* __restrict__ edges, const int* __restrict__ cnt,
                          const float* __restrict__ ev, const float* __restrict__ z,
                          const float* __restrict__ h, float* __restrict__ oacc,
                          int OUT) {
    int gw   = (blockIdx.x * blockDim.x + threadIdx.x) >> 5;
    int lane = threadIdx.x & 31;
    if (gw >= *cnt) return;
    int2 e = edges[gw];
    float alpha = ev[gw] / z[e.y];
    const float* __restrict__ hs = h + (size_t)e.x * OUT;
    float* __restrict__ od = oacc + (size_t)e.y * OUT;
    for (int f = lane; f < OUT; f += 32) atomicAdd(&od[f], alpha * hs[f]);
}

// ---------------------------------------------------------------------------
// 8) y[i-1] = elu( sum_k elu(out[0,k]+b) * elu(out[i,k]+b) ), wave per row i
// ---------------------------------------------------------------------------
__device__ __forceinline__ float eluf(float x) { return x > 0.f ? x : (expf(x) - 1.f); }

__launch_bounds__(256)
__global__ void k_final(const float* __restrict__ oacc, const float* __restrict__ bias,
                        float* __restrict__ y, int item_len, int OUT) {
    int gw   = (blockIdx.x * blockDim.x + threadIdx.x) >> 5;
    int lane = threadIdx.x & 31;
    int i = gw + 1;
    if (i >= item_len) return;
    const float* __restrict__ r0 = oacc;
    const float* __restrict__ ri = oacc + (size_t)i * OUT;
    float s = 0.f;
    for (int k = lane; k < OUT; k += 32) {
        float b  = bias[k];
        float x0 = eluf(r0[k] + b);
        float xi = eluf(ri[k] + b);
        s += x0 * xi;
    }
    #pragma unroll
    for (int o = 16; o > 0; o >>= 1) s += __shfl_xor(s, o, 32);
    if (lane == 0) y[i - 1] = eluf(s);
}

// ---------------------------------------------------------------------------
extern "C" void kernel_launch(void* const* d_in, const int* in_sizes, int n_in,
                              void* d_out, int out_size, void* d_ws, size_t ws_size,
                              hipStream_t stream) {
    const float*     x       = (const float*)d_in[0];
    const long long* ei      = (const long long*)d_in[1];
    const float*     W       = (const float*)d_in[2];
    const float*     att_src = (const float*)d_in[3];
    const float*     att_dst = (const float*)d_in[4];
    const float*     bias    = (const float*)d_in[5];
    float*           y       = (float*)d_out;

    const int OUT      = in_sizes[3];            // att_src length
    const int IN       = in_sizes[2] / OUT;      // W is IN*OUT
    const int N        = in_sizes[0] / IN;       // x is N*IN
    const int E        = in_sizes[1] / 2;        // edge_index is 2*E
    const int item_len = out_size + 1;           // y has item_len-1 entries
    const int M        = E + item_len;           // candidate edges incl. self loops

    // carve workspace (256B aligned slices)
    char* p = (char*)d_ws;
    auto carve = [&](size_t bytes) -> char* {
        char* r = p;
        p += (bytes + 255) & ~(size_t)255;
        return r;
    };
    float*    h     = (float*)carve((size_t)N * OUT * sizeof(float));
    float*    a_s   = (float*)carve((size_t)N * sizeof(float));
    float*    a_d   = (float*)carve((size_t)N * sizeof(float));
    unsigned* mb    = (unsigned*)carve((size_t)item_len * sizeof(unsigned));
    float*    z     = (float*)carve((size_t)item_len * sizeof(float));
    float*    oacc  = (float*)carve((size_t)item_len * OUT * sizeof(float));
    int*      cnt   = (int*)carve(256);
    int2*     edges = (int2*)carve((size_t)M * sizeof(int2));
    float*    ev    = (float*)carve((size_t)M * sizeof(float));
    (void)ws_size;

    // 1) dense projection h = x @ W (fp32 WMMA)
    k_gemm<<<dim3((N + 15) / 16), dim3(256), 0, stream>>>(x, W, h, N, IN, OUT);
    // 2) per-node attention logits
    k_att<<<dim3((N + 7) / 8), dim3(256), 0, stream>>>(h, att_src, att_dst, a_s, a_d, N, OUT);
    // 3) init accumulators / counter
    int initN = item_len * OUT;
    k_init<<<dim3((initN + 255) / 256), dim3(256), 0, stream>>>(mb, z, oacc, cnt, item_len, OUT);
    // 4) compact relevant edges (dst < item_len)
    k_compact<<<dim3((M + 255) / 256), dim3(256), 0, stream>>>(ei, E, item_len, M, edges, cnt);
    // 5..7) segment softmax + weighted scatter on the compacted list
    k_max<<<dim3((M + 255) / 256), dim3(256), 0, stream>>>(edges, cnt, a_s, a_d, ev, mb);
    k_expsum<<<dim3((M + 255) / 256), dim3(256), 0, stream>>>(edges, cnt, ev, mb, z);
    k_scatter<<<dim3((M + 7) / 8), dim3(256), 0, stream>>>(edges, cnt, ev, z, h, oacc, OUT);
    // 8) final elu-dot against row 0
    k_final<<<dim3((item_len + 7) / 8), dim3(256), 0, stream>>>(oacc, bias, y, item_len, OUT);
}